// CapsDecorelationNormalization_89060441849995
// MI455X (gfx1250) — compile-verified
//
#include <hip/hip_runtime.h>

typedef __attribute__((ext_vector_type(2))) float v2f;
typedef __attribute__((ext_vector_type(8))) float v8f;

#define BATCH 2048
#define CI 32
#define CO 32
#define ATOM 16
#define NTOT (BATCH * CO)      // 65536 samples per capsule
#define PBLK 64                // blocks per capsule along batch
#define BPB (BATCH / PBLK)     // 32 b-values per block
#define WAVES 8                // 256 threads = 8 wave32
#define PART_STRIDE 272        // 256 packed sigma + 16 row-sums

static __device__ __forceinline__ v8f wmma_f32_16x16x4(v2f a, v2f b, v8f c) {
  // D(16x16,f32) += A(16x4,f32) * B(4x16,f32)
  return __builtin_amdgcn_wmma_f32_16x16x4_f32(
      /*neg_a=*/false, a, /*neg_b=*/false, b,
      /*c_mod=*/(short)0, c, /*reuse_a=*/false, /*reuse_b=*/false);
}

// -------------------------------------------------------------------------
// Pass 1: per-capsule raw second moment  S2 = X^T X  and row sum, via WMMA.
// For X^T X the A-operand (16x4, M=a-index) and B-operand (4x16, N=a-index)
// have identical per-lane register images under the gfx1250 f32 layouts:
//   lane l<16 : {K=k0+0, K=k0+1} of a-index l ; lane l>=16 : {K=k0+2, K=k0+3}
// so one v2f feeds both operands.
// -------------------------------------------------------------------------
__global__ void caps_stats_kernel(const float* __restrict__ x,
                                  float* __restrict__ partials) {
  const int p    = blockIdx.x;          // batch slab
  const int ci   = blockIdx.y;          // capsule
  const int wave = threadIdx.x >> 5;
  const int lane = threadIdx.x & 31;
  const int half = lane >> 4;
  const int aidx = lane & 15;

  v8f acc = {0.f, 0.f, 0.f, 0.f, 0.f, 0.f, 0.f, 0.f};
  float ssum = 0.f;

  for (int bi = 0; bi < BPB / WAVES; ++bi) {
    const int b = p * BPB + wave * (BPB / WAVES) + bi;
    const float* base = x + ((size_t)b * CI + ci) * (CO * ATOM);
#pragma unroll
    for (int co0 = 0; co0 < CO; co0 += 4) {
      // samples co0..co0+3 form one K=4 chunk
      const float* rp = base + (co0 + 2 * half) * ATOM + aidx;
      v2f ab;
      ab.x = rp[0];       // sample (co0 + 2*half)
      ab.y = rp[ATOM];    // sample (co0 + 2*half + 1)
      acc = wmma_f32_16x16x4(ab, ab, acc);   // S2 += chunk^T * chunk
      ssum += ab.x + ab.y;
    }
  }

  // deterministic in-block reduction over the 8 waves
  __shared__ float lsig[WAVES][256];
  __shared__ float lsum[WAVES][32];
#pragma unroll
  for (int r = 0; r < 8; ++r) lsig[wave][r * 32 + lane] = acc[r];
  lsum[wave][lane] = ssum;
  __syncthreads();

  float* outp = partials + ((size_t)ci * PBLK + p) * PART_STRIDE;
  float s = 0.f;
#pragma unroll
  for (int w = 0; w < WAVES; ++w) s += lsig[w][threadIdx.x];
  outp[threadIdx.x] = s;  // packed in native WMMA C/D order (r*32+lane)
  if (threadIdx.x < 16) {
    float t = 0.f;
#pragma unroll
    for (int w = 0; w < WAVES; ++w)
      t += lsum[w][threadIdx.x] + lsum[w][threadIdx.x + 16];
    outp[256 + threadIdx.x] = t;
  }
}

// -------------------------------------------------------------------------
// Pass 2: one block per capsule. Reduce partials, build sigma, Newton-Schulz
// inverse sqrt (16x16, trivial FLOPs -> LDS matmuls), fold gamma/beta/mean
// into an affine map (Wg, bias). Wg is stored pre-swizzled in the exact
// B-operand lane layout used by pass 3 (8 contiguous floats per lane).
// -------------------------------------------------------------------------
__global__ void caps_solve_kernel(const float* __restrict__ partials,
                                  const float* __restrict__ gamma,
                                  const float* __restrict__ beta,
                                  float* __restrict__ wpk) {
  const int ci = blockIdx.x;
  const int t  = threadIdx.x;

  __shared__ float S[16][17];
  __shared__ float Sn[16][17];
  __shared__ float Pm[16][17];
  __shared__ float T1[16][17];
  __shared__ float T2[16][17];
  __shared__ float meanv[16];
  __shared__ float trsh;

  const float* base = partials + (size_t)ci * PBLK * PART_STRIDE;
  float s2 = 0.f;
  for (int p = 0; p < PBLK; ++p) s2 += base[p * PART_STRIDE + t];
  {
    // unpack WMMA C/D packing: t = r*32 + l ; m = (l<16)? r : r+8 ; n = l&15
    const int r = t >> 5, l = t & 31;
    const int m = (l < 16) ? r : r + 8;
    const int n = l & 15;
    S[m][n] = s2;
  }
  if (t < 16) {
    float su = 0.f;
    for (int p = 0; p < PBLK; ++p) su += base[p * PART_STRIDE + 256 + t];
    meanv[t] = su / (float)NTOT;
  }
  __syncthreads();

  const int mm = t >> 4, nn = t & 15;
  float sig = (S[mm][nn] - (float)NTOT * meanv[mm] * meanv[nn]) /
              ((float)NTOT - 1.0f);
  __syncthreads();
  S[mm][nn] = sig;
  __syncthreads();
  if (t == 0) {
    float a = 0.f;
    for (int i = 0; i < 16; ++i) a += S[i][i];
    trsh = a;
  }
  __syncthreads();
  const float tr = trsh;
  Sn[mm][nn] = S[mm][nn] / tr;
  Pm[mm][nn] = (mm == nn) ? 1.0f : 0.0f;
  __syncthreads();

  for (int it = 0; it < 5; ++it) {   // P = 0.5*(3P - P*P*P*Sn)
    float a = 0.f;
#pragma unroll
    for (int k = 0; k < 16; ++k) a += Pm[mm][k] * Pm[k][nn];
    T1[mm][nn] = a;
    __syncthreads();
    a = 0.f;
#pragma unroll
    for (int k = 0; k < 16; ++k) a += T1[mm][k] * Pm[k][nn];
    T2[mm][nn] = a;
    __syncthreads();
    a = 0.f;
#pragma unroll
    for (int k = 0; k < 16; ++k) a += T2[mm][k] * Sn[k][nn];
    const float pn = 1.5f * Pm[mm][nn] - 0.5f * a;
    __syncthreads();
    Pm[mm][nn] = pn;
    __syncthreads();
  }

  // Wg[a][n] = P[a][n]/sqrt(tr) * gamma[n]
  const float wg = Pm[mm][nn] * rsqrtf(tr) * gamma[ci * ATOM + nn];
  __syncthreads();
  T1[mm][nn] = wg;
  __syncthreads();

  float* wo = wpk + (size_t)ci * PART_STRIDE;
  {
    // B-operand layout for pass 3: wo[l*8 + kk*2 + j] = Wg[kk*4 + 2*(l>>4) + j][l&15]
    const int l = t >> 3, c = t & 7, kk = c >> 1, j = c & 1;
    wo[t] = T1[kk * 4 + 2 * (l >> 4) + j][l & 15];
  }
  if (t < 16) {
    float bsum = 0.f;
#pragma unroll
    for (int a2 = 0; a2 < 16; ++a2) bsum += meanv[a2] * T1[a2][t];
    wo[256 + t] = beta[ci * ATOM + t] - bsum;  // bias[n]
  }
}

// -------------------------------------------------------------------------
// Pass 3: out-tile(16x16) = X-tile(16x16) @ Wg + bias, 4 chained K=4 WMMAs.
// Wg stays resident in registers; accumulator pre-seeded with bias broadcast.
// x re-read hits L2 (128 MB < 192 MB L2).
// -------------------------------------------------------------------------
__global__ void caps_apply_kernel(const float* __restrict__ x,
                                  const float* __restrict__ wpk,
                                  float* __restrict__ out) {
  const int p    = blockIdx.x;
  const int ci   = blockIdx.y;
  const int wave = threadIdx.x >> 5;
  const int lane = threadIdx.x & 31;

  const float* wb = wpk + (size_t)ci * PART_STRIDE;
  const v8f wv = *((const v8f*)wb + lane);   // 32B contiguous per lane
  v2f B0, B1, B2, B3;
  B0.x = wv[0]; B0.y = wv[1];
  B1.x = wv[2]; B1.y = wv[3];
  B2.x = wv[4]; B2.y = wv[5];
  B3.x = wv[6]; B3.y = wv[7];
  const float bias = wb[256 + (lane & 15)];
  v8f cinit;
#pragma unroll
  for (int r = 0; r < 8; ++r) cinit[r] = bias;

  const int row  = lane & 15;
  const int koff = 2 * (lane >> 4);
  const int mrow = (lane < 16) ? 0 : 8;

  for (int bi = 0; bi < BPB / WAVES; ++bi) {
    const int b = p * BPB + wave * (BPB / WAVES) + bi;
    const size_t off = ((size_t)b * CI + ci) * (CO * ATOM);
    const float* base = x + off;
    float* obase = out + off;
#pragma unroll
    for (int t0 = 0; t0 < CO / 16; ++t0) {
      const float* rp = base + (t0 * 16 + row) * ATOM + koff;
      v2f A0, A1, A2, A3;            // per-lane b64 loads (2 consecutive f32)
      A0.x = rp[0];  A0.y = rp[1];
      A1.x = rp[4];  A1.y = rp[5];
      A2.x = rp[8];  A2.y = rp[9];
      A3.x = rp[12]; A3.y = rp[13];
      v8f d = cinit;
      d = wmma_f32_16x16x4(A0, B0, d);
      d = wmma_f32_16x16x4(A1, B1, d);
      d = wmma_f32_16x16x4(A2, B2, d);
      d = wmma_f32_16x16x4(A3, B3, d);
      float* ob = obase + t0 * 16 * ATOM;
#pragma unroll
      for (int r = 0; r < 8; ++r)
        ob[(r + mrow) * ATOM + (lane & 15)] = d[r];
    }
  }
}

extern "C" void kernel_launch(void* const* d_in, const int* in_sizes, int n_in,
                              void* d_out, int out_size, void* d_ws, size_t ws_size,
                              hipStream_t stream) {
  const float* x     = (const float*)d_in[0];
  const float* gamma = (const float*)d_in[1];
  const float* beta  = (const float*)d_in[2];
  float* out = (float*)d_out;

  float* partials = (float*)d_ws;                               // CI*PBLK*272 f32
  float* wpk = partials + (size_t)CI * PBLK * PART_STRIDE;      // CI*272 f32

  dim3 grid(PBLK, CI);
  caps_stats_kernel<<<grid, WAVES * 32, 0, stream>>>(x, partials);
  caps_solve_kernel<<<CI, 256, 0, stream>>>(partials, gamma, beta, wpk);
  caps_apply_kernel<<<grid, WAVES * 32, 0, stream>>>(x, wpk, out);
}